// DMLoss_69320772157502
// MI455X (gfx1250) — compile-verified
//
#include <hip/hip_runtime.h>

typedef __attribute__((ext_vector_type(2))) float v2f;
typedef __attribute__((ext_vector_type(8))) float v8f;

#define NPTS 128
#define MINTERP 1280   // NPTS * INTERP(10)
#define TPB 256

__device__ __forceinline__ float sl1(float d) {
    float a = fabsf(d);
    return a < 1.0f ? 0.5f * a * a : a - 0.5f;   // SMOOTH_L1_BETA = 1.0
}

// Tree argmin over the 8 accumulator rows of one WMMA tile.
// Tie-break keeps the LEFT operand, which always has the smaller local index,
// and the running compare is strict '<', so first-occurrence argmin semantics hold.
__device__ __forceinline__ void argmin8(const v8f& d, int rowbase,
                                        float& minv, int& mini) {
    bool c;
    float m0, m1, m2, m3; int i0, i1, i2, i3;
    c = d[1] < d[0]; m0 = c ? d[1] : d[0]; i0 = c ? 1 : 0;
    c = d[3] < d[2]; m1 = c ? d[3] : d[2]; i1 = c ? 3 : 2;
    c = d[5] < d[4]; m2 = c ? d[5] : d[4]; i2 = c ? 5 : 4;
    c = d[7] < d[6]; m3 = c ? d[7] : d[6]; i3 = c ? 7 : 6;
    c = m1 < m0;     m0 = c ? m1 : m0;     i0 = c ? i1 : i0;
    c = m3 < m2;     m2 = c ? m3 : m2;     i2 = c ? i3 : i2;
    c = m2 < m0;     m0 = c ? m2 : m0;     i0 = c ? i2 : i0;
    int gi = rowbase + i0;
    c = m0 < minv;
    minv = c ? m0 : minv;
    mini = c ? gi : mini;
}

__global__ void dmloss_main(const float* __restrict__ ini,
                            const float* __restrict__ pred,
                            const float* __restrict__ gt,
                            const float* __restrict__ kmask,
                            float* __restrict__ ws) {
    // Pre-scaled A-fragment operands: one ds_load_b64 per tile per lane.
    __shared__ v2f I2[MINTERP];    // {-2*ix, -2*iy}
    __shared__ v2f G2p[MINTERP];   // {g2, 0}
    __shared__ v2f PA[NPTS];       // {-2*px, -2*py}   (ini_pred)
    __shared__ v2f P2p[NPTS];      // {p2, 0}          (ini_pred)
    __shared__ float PPx[NPTS], PPy[NPTS];   // pred_poly (raw)
    __shared__ float Gx[NPTS], Gy[NPTS], Mk[NPTS];
    __shared__ float red[3 * TPB];

    const int tid  = threadIdx.x;
    const int b    = blockIdx.x;
    const int lane = tid & 31;
    const int wv   = tid >> 5;        // 8 waves, one 16-wide column tile each
    const int l16  = lane & 15;
    const bool lo  = (lane < 16);

    const float* gtb = gt   + (size_t)b * NPTS * 2;
    const float* inb = ini  + (size_t)b * NPTS * 2;
    const float* prb = pred + (size_t)b * NPTS * 2;
    const float* mkb = kmask + (size_t)b * NPTS;

    // ---------------- Phase 1: stage inputs
    if (tid < NPTS) {
        float gx = gtb[2*tid], gy = gtb[2*tid+1];
        Gx[tid] = gx; Gy[tid] = gy;
        float px = inb[2*tid], py = inb[2*tid+1];
        v2f pa; pa.x = -2.0f*px; pa.y = -2.0f*py; PA[tid] = pa;
        v2f pp; pp.x = px*px + py*py; pp.y = 0.0f; P2p[tid] = pp;
        PPx[tid] = prb[2*tid]; PPy[tid] = prb[2*tid+1];
        Mk[tid]  = mkb[tid];
    }
    __syncthreads();
    // Build interpolated GT polygon (pre-scaled)
    for (int t = tid; t < MINTERP; t += TPB) {
        int m  = t / 10;
        int j  = t - 10*m;
        int mp = (m + NPTS - 1) & (NPTS - 1);   // roll(+1): previous vertex
        float s  = (float)j * 0.1f;
        float os = 1.0f - s;
        float x = Gx[m]*s + Gx[mp]*os;
        float y = Gy[m]*s + Gy[mp]*os;
        v2f iv; iv.x = -2.0f*x; iv.y = -2.0f*y; I2[t] = iv;
        v2f gv; gv.x = x*x + y*y; gv.y = 0.0f;  G2p[t] = gv;
    }
    __syncthreads();

    // ---------------- d1: score[m,n] = g2[m] - 2*gp ; argmin over m (1280 interp rows)
    // A row m: (-2gx, -2gy, g2, 0) ; B col n: (px, py, 1, 0) ; V_WMMA_F32_16X16X4_F32
    const int ncol = 16*wv + l16;
    v2f bfr;
    {
        v2f pa = PA[ncol];
        bfr.x = lo ? (-0.5f * pa.x) : 1.0f;   // lanes<16: K0/K1 ; lanes>=16: K2/K3
        bfr.y = lo ? (-0.5f * pa.y) : 0.0f;
    }
    float minv = 3.4e38f; int mini = 0;
    {
        const v2f* __restrict__ ap = lo ? I2 : G2p;   // lane-selected base
        v2f acur = ap[l16];
        for (int r = 0; r < MINTERP/16; ++r) {
            int rn = (r + 1 < MINTERP/16) ? (r + 1) : r;
            v2f anext = ap[16*rn + l16];              // prefetch next tile
            v8f c = {};
            v8f dacc = __builtin_amdgcn_wmma_f32_16x16x4_f32(
                false, acur, false, bfr, (short)0, c, false, false);
            argmin8(dacc, 16*r + (lo ? 0 : 8), minv, mini);
            acur = anext;
        }
    }
    {   // merge the two half-wave row groups (M=v vs M=8+v)
        float ov = __shfl_xor(minv, 16, 32);
        int   oi = __shfl_xor(mini, 16, 32);
        if (ov < minv || (ov == minv && oi < mini)) { minv = ov; mini = oi; }
    }
    float p_pred2gt = 0.0f;
    if (lo) {
        v2f iv = I2[mini];
        float ngx = -0.5f * iv.x, ngy = -0.5f * iv.y;
        p_pred2gt = sl1(PPx[ncol] - ngx) + sl1(PPy[ncol] - ngy);
    }

    // ---------------- d2 (transposed): score[n,m] = p2[n] - 2*gp ; argmin over n
    const int mcol = 16*wv + l16;
    v2f bfr2;
    bfr2.x = lo ? Gx[mcol] : 1.0f;
    bfr2.y = lo ? Gy[mcol] : 0.0f;
    float minv2 = 3.4e38f; int mini2 = 0;
    {
        const v2f* __restrict__ ap = lo ? PA : P2p;
        v2f acur = ap[l16];
        for (int r = 0; r < NPTS/16; ++r) {
            int rn = (r + 1 < NPTS/16) ? (r + 1) : r;
            v2f anext = ap[16*rn + l16];
            v8f c = {};
            v8f dacc = __builtin_amdgcn_wmma_f32_16x16x4_f32(
                false, acur, false, bfr2, (short)0, c, false, false);
            argmin8(dacc, 16*r + (lo ? 0 : 8), minv2, mini2);
            acur = anext;
        }
    }
    {
        float ov = __shfl_xor(minv2, 16, 32);
        int   oi = __shfl_xor(mini2, 16, 32);
        if (ov < minv2 || (ov == minv2 && oi < mini2)) { minv2 = ov; mini2 = oi; }
    }
    float p_gt2pred = 0.0f, p_mask = 0.0f;
    if (lo) {
        float npx = PPx[mini2], npy = PPy[mini2];
        float mm = Mk[mcol];
        p_gt2pred = (sl1(npx - Gx[mcol]) + sl1(npy - Gy[mcol])) * mm;
        p_mask = mm;
    }

    // ---------------- fused 3-channel block reduction (deterministic)
    red[tid]           = p_pred2gt;
    red[TPB + tid]     = p_gt2pred;
    red[2*TPB + tid]   = p_mask;
    __syncthreads();
    for (int s = TPB/2; s > 0; s >>= 1) {
        if (tid < s) {
            red[tid]         += red[tid + s];
            red[TPB + tid]   += red[TPB + tid + s];
            red[2*TPB + tid] += red[2*TPB + tid + s];
        }
        __syncthreads();
    }
    if (tid == 0) {
        ws[3*b + 0] = red[0];
        ws[3*b + 1] = red[TPB];
        ws[3*b + 2] = red[2*TPB];
    }
}

__global__ void dmloss_final(const float* __restrict__ ws, float* __restrict__ out, int B) {
    __shared__ float red[3 * TPB];
    float s0 = 0.f, s1 = 0.f, s2 = 0.f;
    for (int b = threadIdx.x; b < B; b += TPB) {
        s0 += ws[3*b + 0];
        s1 += ws[3*b + 1];
        s2 += ws[3*b + 2];
    }
    red[threadIdx.x] = s0; red[TPB + threadIdx.x] = s1; red[2*TPB + threadIdx.x] = s2;
    __syncthreads();
    for (int s = TPB/2; s > 0; s >>= 1) {
        if (threadIdx.x < s) {
            red[threadIdx.x]         += red[threadIdx.x + s];
            red[TPB + threadIdx.x]   += red[TPB + threadIdx.x + s];
            red[2*TPB + threadIdx.x] += red[2*TPB + threadIdx.x + s];
        }
        __syncthreads();
    }
    if (threadIdx.x == 0) {
        float pred2gt    = red[0] / ((float)B * (float)NPTS * 2.0f); // mean over (B,N,2)
        float masked_sum = red[TPB];
        float mask_sum   = 2.0f * red[2*TPB];
        float loss_set2set = masked_sum / (mask_sum + 1.0f) + pred2gt;
        out[0] = 1.0f * (loss_set2set * 0.5f);    // LOSS_WEIGHT * loss
    }
}

extern "C" void kernel_launch(void* const* d_in, const int* in_sizes, int n_in,
                              void* d_out, int out_size, void* d_ws, size_t ws_size,
                              hipStream_t stream) {
    const float* ini  = (const float*)d_in[0];   // ini_pred_poly (B,128,2)
    const float* pred = (const float*)d_in[1];   // pred_poly     (B,128,2)
    const float* gt   = (const float*)d_in[2];   // gt_poly       (B,128,2)
    const float* km   = (const float*)d_in[3];   // keyPointsMask (B,128)
    int B = in_sizes[0] / (NPTS * 2);
    float* ws = (float*)d_ws;                    // B*3 floats of partials

    dmloss_main<<<B, TPB, 0, stream>>>(ini, pred, gt, km, ws);
    dmloss_final<<<1, TPB, 0, stream>>>(ws, (float*)d_out, B);
}